// GPUPreprocessor_83056077570234
// MI455X (gfx1250) — compile-verified
//
#include <hip/hip_runtime.h>
#include <stdint.h>

// Preprocess: [16,1280,960,3] f32 -> rescale(1/255) -> bilinear 640x640
// (half-pixel centers) -> normalize -> NCHW [16,3,640,640].
//
// Memory-bound (~315 MB moved, ~13.5us at 23.3 TB/s). WMMA is inappropriate
// (dense-matmul formulation wastes ~500x FLOPs for no byte savings); the
// CDNA5 feature that matters is the async global->LDS datamover
// (GLOBAL_LOAD_ASYNC_TO_LDS_B128, ASYNCcnt). SADDR form keeps the uniform
// chunk base in SGPRs (no per-lane 64-bit address math), and th:TH_LOAD_NT
// keeps the single-use input stream from evicting useful L2 lines.

#define IN_H   1280
#define IN_W   960
#define OUT_H  640
#define OUT_W  640
#define ROWF   (IN_W * 3)        // floats per input row (2880)
#define COPYF  (2 * ROWF)        // floats staged per block (5760)
#define COPYB  (COPYF * 4)       // 23040 bytes, 16B-multiple
#define NXFER  (COPYB / 16)      // 1440 b128 lane transfers
#define NTHR   320               // 10 wave32s; 640 out px = 2 per thread

__global__ __launch_bounds__(NTHR)
void GPUPreprocessor_83056077570234_kernel(const float* __restrict__ img,
                                           const float* __restrict__ mean,
                                           const float* __restrict__ stdv,
                                           float* __restrict__ out)
{
    __shared__ __align__(16) float smem[COPYF];

    const int y   = blockIdx.x;   // output row   0..639
    const int b   = blockIdx.y;   // batch        0..15
    const int tid = threadIdx.x;  // 0..319

    // ---- Stage input rows (2y, 2y+1) — one contiguous 23040B chunk — into
    // ---- LDS via the async datamover. SADDR form: uniform base in SGPRs,
    // ---- per-lane 32-bit byte offset. 1440 b128 transfers = 4 full issues
    // ---- of 320 lanes + 1 masked issue (tid < 160).
    const uint64_t gbase =
        (uint64_t)(uintptr_t)(img + (size_t)(b * IN_H + 2 * y) * ROWF);
    const uint32_t lbase = (uint32_t)(uintptr_t)(&smem[0]); // low 32b = LDS addr
    const uint32_t boff0 = (uint32_t)tid * 16u;             // lane byte offset

    #pragma unroll
    for (int i = 0; i < 4; ++i) {
        uint32_t off = boff0 + (uint32_t)i * (NTHR * 16u);
        uint32_t l   = lbase + off;
        asm volatile("global_load_async_to_lds_b128 %0, %1, %2 th:TH_LOAD_NT"
                     :: "v"(l), "v"(off), "s"(gbase)
                     : "memory");
    }
    if (tid < (NXFER - 4 * NTHR)) {               // tail: 160 lanes
        uint32_t off = boff0 + 4u * (NTHR * 16u);
        uint32_t l   = lbase + off;
        asm volatile("global_load_async_to_lds_b128 %0, %1, %2 th:TH_LOAD_NT"
                     :: "v"(l), "v"(off), "s"(gbase)
                     : "memory");
    }
    asm volatile("s_wait_asynccnt 0" ::: "memory"); // this wave's LDS writes done
    __syncthreads();                                // all waves' chunks visible

    // ---- Fold rescale+normalize into one fma: (v/255 - m)/s = v*sc + bs
    float sc[3], bs[3];
    #pragma unroll
    for (int c = 0; c < 3; ++c) {
        const float s = stdv[c];
        sc[c] = (1.0f / 255.0f) / s;
        bs[c] = -mean[c] / s;
    }

    // ---- Horizontal (scale 1.5, half-pixel): output pair (2k,2k+1) taps
    // ---- input columns 3k..3k+2 with fx = 0.25 / 0.75. Vertical (scale
    // ---- 2.0): rows 2y,2y+1 averaged 0.5/0.5 — no edge clamps anywhere.
    const float* r0 = smem;          // input row 2y
    const float* r1 = smem + ROWF;   // input row 2y+1
    const int base = tid * 9;        // col (3*tid), 3 ch, 3 cols -> 9 floats

    float a0[3], a1[3], a2[3];       // row-summed taps (vertical avg deferred)
    #pragma unroll
    for (int c = 0; c < 3; ++c) {
        a0[c] = r0[base + c]     + r1[base + c];
        a1[c] = r0[base + 3 + c] + r1[base + 3 + c];
        a2[c] = r0[base + 6 + c] + r1[base + 6 + c];
    }

    #pragma unroll
    for (int c = 0; c < 3; ++c) {
        // 0.5 vertical weight folded into the horizontal lerp coefficients
        const float ve = 0.375f * a0[c] + 0.125f * a1[c]; // x=2k   (fx=0.25)
        const float vo = 0.125f * a1[c] + 0.375f * a2[c]; // x=2k+1 (fx=0.75)
        float2 o;
        o.x = fmaf(ve, sc[c], bs[c]);
        o.y = fmaf(vo, sc[c], bs[c]);
        // NCHW: out[b][c][y][2*tid .. 2*tid+1], coalesced float2 (b64) store
        float2* dst =
            (float2*)(out + ((size_t)(b * 3 + c) * OUT_H + y) * OUT_W) + tid;
        *dst = o;
    }
}

extern "C" void kernel_launch(void* const* d_in, const int* in_sizes, int n_in,
                              void* d_out, int out_size, void* d_ws, size_t ws_size,
                              hipStream_t stream)
{
    (void)in_sizes; (void)n_in; (void)out_size; (void)d_ws; (void)ws_size;
    const float* img  = (const float*)d_in[0]; // [16,1280,960,3]
    const float* mean = (const float*)d_in[1]; // [3]
    const float* stdv = (const float*)d_in[2]; // [3]
    float*       out  = (float*)d_out;         // [16,3,640,640]

    dim3 grid(OUT_H, 16);   // one block per (output row, batch)
    GPUPreprocessor_83056077570234_kernel<<<grid, NTHR, 0, stream>>>(
        img, mean, stdv, out);
}